// TriangleAttention_21225728377412
// MI455X (gfx1250) — compile-verified
//
#include <hip/hip_runtime.h>

// ---------------------------------------------------------------------------
// Triangle attention for MI455X (gfx1250), f32 end-to-end using
// V_WMMA_F32_16X16X4_F32 (native f32 matrix path, wave32).
// B=1, N=256, D=128, H=4, DH=32.
// ---------------------------------------------------------------------------

typedef float v2f __attribute__((ext_vector_type(2)));
typedef float v8f __attribute__((ext_vector_type(8)));

#define WMMA_F32(a, b, c)                                                     \
  __builtin_amdgcn_wmma_f32_16x16x4_f32(false, (a), false, (b), (short)0,     \
                                        (c), false, false)

#define NSEQ 256
#define DDIM 128
#define NH 4
#define DH 32

// LDS strides (floats), padded to avoid bank conflicts on 64x4B banks.
#define QKV_STR 36   // 144B rows: 8B aligned for b64 ds loads, spreads banks
#define P_STR 18     // 72B rows: 8B aligned
#define WO_STR 136   // 2*136 % 64 == 16 -> half-waves hit disjoint bank ranges

// ---------------------------------------------------------------------------
// Kernel 1: attn_bias[h][i][j] = sum_d pairwise[i,j,d] * w_bias[d,h]
// ---------------------------------------------------------------------------
__global__ void __launch_bounds__(256) bias_kernel(
    const float* __restrict__ pr, const float* __restrict__ wb,
    float* __restrict__ bias) {
  const int i = blockIdx.x;
  const int j = threadIdx.x;
  const float4* row = (const float4*)(pr + ((size_t)i * NSEQ + j) * DDIM);
  float a0 = 0.f, a1 = 0.f, a2 = 0.f, a3 = 0.f;
#pragma unroll 8
  for (int d4 = 0; d4 < DDIM / 4; ++d4) {
    float4 x = row[d4];
    const float4* w = (const float4*)(wb + d4 * 16);
    float4 w0 = w[0], w1 = w[1], w2 = w[2], w3 = w[3];
    a0 += x.x * w0.x + x.y * w1.x + x.z * w2.x + x.w * w3.x;
    a1 += x.x * w0.y + x.y * w1.y + x.z * w2.y + x.w * w3.y;
    a2 += x.x * w0.z + x.y * w1.z + x.z * w2.z + x.w * w3.z;
    a3 += x.x * w0.w + x.y * w1.w + x.z * w2.w + x.w * w3.w;
  }
  const int ij = i * NSEQ + j;
  bias[0 * NSEQ * NSEQ + ij] = a0;
  bias[1 * NSEQ * NSEQ + ij] = a1;
  bias[2 * NSEQ * NSEQ + ij] = a2;
  bias[3 * NSEQ * NSEQ + ij] = a3;
}

// ---------------------------------------------------------------------------
// Kernel 2: per (row-batch rb, head h): QKV projection + flash attention.
// grid = 1024 (rb*4+h), block = 256 threads = 8 waves.
// ---------------------------------------------------------------------------
__global__ void __launch_bounds__(256, 1) attn_kernel(
    const float* __restrict__ pr, const float* __restrict__ wq,
    const float* __restrict__ wk, const float* __restrict__ wv,
    const float* __restrict__ bias, float* __restrict__ ctx) {
  extern __shared__ float sm[];
  float* qs = sm;                               // 256*36
  float* ks = qs + NSEQ * QKV_STR;              // 256*36
  float* vs = ks + NSEQ * QKV_STR;              // 256*36
  float* ps = vs + NSEQ * QKV_STR;              // 8 * 16*18

  const int rb = blockIdx.x >> 2;
  const int h = blockIdx.x & 3;
  const int tid = threadIdx.x;
  const int w = tid >> 5;       // wave id 0..7
  const int l = tid & 31;       // lane
  const int mrow = l & 15;      // row/col within 16-lane group
  const int lh = l >> 4;        // half-wave select (WMMA K/M split)

  const float* x = pr + (size_t)rb * NSEQ * DDIM;

  // ---- Phase 1: Q/K/V = x[256,128] @ W[:, h*32 : h*32+32] via f32 WMMA ----
  // Wave w computes rows [32w, 32w+32): 2 M-tiles. Per K-step, one shared
  // x A-fragment feeds 6 accumulators (q/k/v x 2 N-tiles) = 6 WMMAs.
  // All pointers are compile-time-known address spaces: global loads for
  // weights, ds stores for LDS staging (no FLAT ops).
  for (int half = 0; half < 2; ++half) {
    const int mt = w * 2 + half;
    const float* Arow = x + (mt * 16 + mrow) * DDIM + lh * 2;
    v8f cq0 = {}, cq1 = {}, ck0 = {}, ck1 = {}, cv0 = {}, cv1 = {};
#pragma unroll 4
    for (int k0 = 0; k0 < DDIM; k0 += 4) {
      v2f a;
      a.x = Arow[k0];
      a.y = Arow[k0 + 1];
      const int ro = (k0 + lh * 2) * DDIM + h * DH + mrow;
      v2f b;
      b.x = wq[ro];        b.y = wq[ro + DDIM];        cq0 = WMMA_F32(a, b, cq0);
      b.x = wq[ro + 16];   b.y = wq[ro + DDIM + 16];   cq1 = WMMA_F32(a, b, cq1);
      b.x = wk[ro];        b.y = wk[ro + DDIM];        ck0 = WMMA_F32(a, b, ck0);
      b.x = wk[ro + 16];   b.y = wk[ro + DDIM + 16];   ck1 = WMMA_F32(a, b, ck1);
      b.x = wv[ro];        b.y = wv[ro + DDIM];        cv0 = WMMA_F32(a, b, cv0);
      b.x = wv[ro + 16];   b.y = wv[ro + DDIM + 16];   cv1 = WMMA_F32(a, b, cv1);
    }
#pragma unroll
    for (int r = 0; r < 8; ++r) {
      const int row = (mt * 16 + r + lh * 8) * QKV_STR;
      qs[row + mrow] = cq0[r];
      qs[row + 16 + mrow] = cq1[r];
      ks[row + mrow] = ck0[r];
      ks[row + 16 + mrow] = ck1[r];
      vs[row + mrow] = cv0[r];
      vs[row + 16 + mrow] = cv1[r];
    }
  }
  __syncthreads();

  // ---- Phase 2: flash attention, wave w owns rows [32w, 32w+32) ----
  const float scale = 0.17677669529663687f;  // 1/sqrt(32)
  float* pscr = ps + w * 16 * P_STR;
  const float* biasH = bias + (size_t)h * NSEQ * NSEQ;

  for (int half = 0; half < 2; ++half) {
    const int mt = w * 2 + half;
    const int m0 = mt * 16;

    // Preload Q A-fragments for this 16-row tile (dh = 32 -> 8 K-steps).
    v2f aq[8];
#pragma unroll
    for (int kk = 0; kk < 8; ++kk) {
      const float* p = qs + (m0 + mrow) * QKV_STR + kk * 4 + lh * 2;
      aq[kk].x = p[0];
      aq[kk].y = p[1];
    }

    float rmax[8], rsum[8];
    v8f o0 = {}, o1 = {};
#pragma unroll
    for (int r = 0; r < 8; ++r) {
      rmax[r] = -3.0e38f;
      rsum[r] = 0.f;
    }

    for (int jt = 0; jt < 16; ++jt) {
      const int j0 = jt * 16;

      // S tile = q @ k^T  (8 WMMAs over dh)
      v8f s = {};
#pragma unroll
      for (int kk = 0; kk < 8; ++kk) {
        const float* p = ks + (j0 + mrow) * QKV_STR + kk * 4 + lh * 2;
        v2f b;
        b.x = p[0];
        b.y = p[1];
        s = WMMA_F32(aq[kk], b, s);
      }

      // scale + bias (bias depends on (i,j) only; L2-resident, reused 256x)
      float sv[8];
#pragma unroll
      for (int r = 0; r < 8; ++r)
        sv[r] = s[r] * scale + biasH[(m0 + r + lh * 8) * NSEQ + j0 + mrow];

      // Row-wise online softmax. Each C row lives in one 16-lane group.
#pragma unroll
      for (int r = 0; r < 8; ++r) {
        float tm = sv[r];
        tm = fmaxf(tm, __shfl_xor(tm, 1));
        tm = fmaxf(tm, __shfl_xor(tm, 2));
        tm = fmaxf(tm, __shfl_xor(tm, 4));
        tm = fmaxf(tm, __shfl_xor(tm, 8));
        const float nm = fmaxf(rmax[r], tm);
        const float alpha = __expf(rmax[r] - nm);
        rmax[r] = nm;
        const float p = __expf(sv[r] - nm);
        float ts = p;
        ts += __shfl_xor(ts, 1);
        ts += __shfl_xor(ts, 2);
        ts += __shfl_xor(ts, 4);
        ts += __shfl_xor(ts, 8);
        rsum[r] = rsum[r] * alpha + ts;
        o0[r] *= alpha;
        o1[r] *= alpha;
        pscr[(r + lh * 8) * P_STR + mrow] = p;  // C-layout -> LDS
      }

      // O += P @ V (re-read P as A-fragments; per-wave LDS is in-order)
#pragma unroll
      for (int kk = 0; kk < 4; ++kk) {
        v2f a;
        const float* pa = pscr + mrow * P_STR + kk * 4 + lh * 2;
        a.x = pa[0];
        a.y = pa[1];
        const float* pb = vs + (j0 + kk * 4 + lh * 2) * QKV_STR;
        v2f b0, b1;
        b0.x = pb[mrow];
        b0.y = pb[QKV_STR + mrow];
        b1.x = pb[16 + mrow];
        b1.y = pb[QKV_STR + 16 + mrow];
        o0 = WMMA_F32(a, b0, o0);
        o1 = WMMA_F32(a, b1, o1);
      }
    }

    // Epilogue: normalize and write ctx[rb*256+i, h*32+dh]
    float* out = ctx + ((size_t)rb * NSEQ + m0) * DDIM + h * DH;
#pragma unroll
    for (int r = 0; r < 8; ++r) {
      const float inv = 1.0f / rsum[r];
      out[(r + lh * 8) * DDIM + mrow] = o0[r] * inv;
      out[(r + lh * 8) * DDIM + 16 + mrow] = o1[r] * inv;
    }
  }
}

// ---------------------------------------------------------------------------
// Kernel 3: out[65536,128] = ctx[65536,128] @ w_o[128,128]. WMMA GEMM.
// grid = 512 (128-row strips), block = 256 (8 waves, each: 1 Mtile x 8 Ntiles)
// ---------------------------------------------------------------------------
__global__ void __launch_bounds__(256, 1) oproj_kernel(
    const float* __restrict__ ctx, const float* __restrict__ wo,
    float* __restrict__ out) {
  extern __shared__ float sw[];  // w_o staged [128][WO_STR]
  const int tid = threadIdx.x;
  {
    const float4* src = (const float4*)wo;
    for (int i4 = tid; i4 < (DDIM * DDIM) / 4; i4 += 256) {
      const int d = i4 >> 5;        // row
      const int c4 = i4 & 31;       // float4 col
      *(float4*)(sw + d * WO_STR + c4 * 4) = src[i4];
    }
  }
  __syncthreads();

  const int row0 = blockIdx.x * 128;
  const int w = tid >> 5, l = tid & 31;
  const int mrow = l & 15, lh = l >> 4;

  const float* Arow = ctx + (size_t)(row0 + w * 16 + mrow) * DDIM + lh * 2;
  v8f c[8] = {};
#pragma unroll 2
  for (int k0 = 0; k0 < DDIM; k0 += 4) {
    v2f a;
    a.x = Arow[k0];
    a.y = Arow[k0 + 1];
    const float* Bb = sw + (k0 + lh * 2) * WO_STR;
#pragma unroll
    for (int nt = 0; nt < 8; ++nt) {
      v2f b;
      b.x = Bb[nt * 16 + mrow];
      b.y = Bb[WO_STR + nt * 16 + mrow];
      c[nt] = WMMA_F32(a, b, c[nt]);
    }
  }

  float* o = out + (size_t)(row0 + w * 16) * DDIM;
#pragma unroll
  for (int nt = 0; nt < 8; ++nt)
#pragma unroll
    for (int r = 0; r < 8; ++r)
      o[(r + lh * 8) * DDIM + nt * 16 + mrow] = c[nt][r];
}

// ---------------------------------------------------------------------------
extern "C" void kernel_launch(void* const* d_in, const int* in_sizes, int n_in,
                              void* d_out, int out_size, void* d_ws,
                              size_t ws_size, hipStream_t stream) {
  const float* pr = (const float*)d_in[0];   // [1,256,256,128]
  const float* wq = (const float*)d_in[1];   // [128,128]
  const float* wk = (const float*)d_in[2];   // [128,128]
  const float* wv = (const float*)d_in[3];   // [128,128]
  const float* wo = (const float*)d_in[4];   // [128,128]
  const float* wb = (const float*)d_in[5];   // [128,4]

  float* bias = (float*)d_ws;                       // 4*256*256 floats (1 MB)
  float* ctx = bias + (size_t)NH * NSEQ * NSEQ;     // 65536*128 floats (32 MB)
  float* out = (float*)d_out;

  bias_kernel<<<NSEQ, NSEQ, 0, stream>>>(pr, wb, bias);

  const size_t attn_lds =
      (size_t)(3 * NSEQ * QKV_STR + 8 * 16 * P_STR) * sizeof(float);
  attn_kernel<<<NSEQ * NH, 256, attn_lds, stream>>>(pr, wq, wk, wv, bias, ctx);

  const size_t oproj_lds = (size_t)DDIM * WO_STR * sizeof(float);
  oproj_kernel<<<(NSEQ * NSEQ) / 128, 256, oproj_lds, stream>>>(ctx, wo, out);
}